// MultiHeadedAttention_28363964022845
// MI455X (gfx1250) — compile-verified
//
#include <hip/hip_runtime.h>
#include <hip/hip_bf16.h>

// ---------------------------------------------------------------------------
// Decode-step MHA (B=8, Q=1, H=32, D=128, HID=4096, S=4096) for gfx1250.
// Memory-bound (~1.34 GB/call -> ~57us floor at 23.3 TB/s). GEMM tiles use
// V_WMMA_F32_16X16X4_F32 (exact fp32 on the matrix pipe); attention is
// flash-decode split 4-ways over the sequence with a log-sum-exp combine.
// Streamed-once data (weights, KV caches) uses non-temporal loads so the
// reusable activations / q vectors stay resident in WGP$/L2.
// ---------------------------------------------------------------------------

typedef __attribute__((ext_vector_type(2))) float v2f;
typedef __attribute__((ext_vector_type(4))) float v4f;
typedef __attribute__((ext_vector_type(8))) float v8f;

#define HID 4096
#define NHEAD 32
#define HDIM 128
#define NB 8
#define SEQ 4096
#define NEWPOS 4095
#define NPART 4
#define TPART 1024

// ----- fp32 WMMA GEMM tile: C[16x16] = Xpad[16x4096] * W[16rows x 4096]^T -----
// A (16x4 f32): lane L holds row M=L%16, k-phase 2*(L>>4): VGPR0=K+0, VGPR1=K+1
// B (4x16 f32): lane L holds col N=L%16, k-phase 2*(L>>4): VGPR0=K+0, VGPR1=K+1
// C (16x16 f32): lane L<16 -> VGPR j = (M=j, N=L); lanes>=16 hold M=8..15 (pad).
__device__ __forceinline__ v8f gemm_tile16(const float* __restrict__ X,
                                           const float* __restrict__ W,
                                           int n0, int lane) {
  const int r   = lane & 15;         // row (m) for A / col (n) for B
  const int kph = (lane >> 4) << 1;  // k phase: 0 or 2
  const float msk = (r < NB) ? 1.0f : 0.0f;   // pad batches 8..15 with zeros
  const float* __restrict__ xrow = X + (size_t)(r & (NB - 1)) * HID + kph;
  const float* __restrict__ wrow = W + (size_t)(n0 + r) * HID + kph;
  v8f c = (v8f)0.0f;
#pragma unroll 8
  for (int k0 = 0; k0 < HID; k0 += 4) {
    v2f a = *(const v2f*)(xrow + k0);          // activations: reused by all
    a *= msk;                                  //   n-tiles -> keep cached (RT)
    v2f b = __builtin_nontemporal_load(        // weights: streamed once -> NT
        (const v2f*)(wrow + k0));
    c = __builtin_amdgcn_wmma_f32_16x16x4_f32(
        /*neg_a=*/false, a, /*neg_b=*/false, b,
        /*c_mod=*/(short)0, c, /*reuse_a=*/false, /*reuse_b=*/false);
  }
  return c;
}

// ----- Kernel 1: fused QKV projection + bias + RoPE -> workspace -------------
// 768 blocks x 32 threads. block = mat(0..2)*256 + n-tile(0..255).
__global__ __launch_bounds__(32) void qkv_rope_kernel(
    const float* __restrict__ query, const float* __restrict__ key_in,
    const float* __restrict__ value_in,
    const float* __restrict__ wq, const float* __restrict__ bq,
    const float* __restrict__ wk, const float* __restrict__ bk,
    const float* __restrict__ wv, const float* __restrict__ bv,
    const float* __restrict__ fcos, const float* __restrict__ fsin,
    float* __restrict__ q_ws, float* __restrict__ k_ws,
    float* __restrict__ v_ws) {
  const int tile = blockIdx.x;
  const int mat  = tile >> 8;              // 0=q 1=k 2=v
  const int n0   = (tile & 255) * 16;
  const int lane = threadIdx.x;

  const float* X  = (mat == 0) ? query : (mat == 1) ? key_in : value_in;
  const float* W  = (mat == 0) ? wq    : (mat == 1) ? wk     : wv;
  const float* Bb = (mat == 0) ? bq    : (mat == 1) ? bk     : bv;
  float*       O  = (mat == 0) ? q_ws  : (mat == 1) ? k_ws   : v_ws;

  v8f c = gemm_tile16(X, W, n0, lane);

  const int n = n0 + (lane & 15);
  const float bias = Bb[n];
#pragma unroll
  for (int j = 0; j < 8; ++j) c[j] += bias;

  if (mat < 2) {  // RoPE on q and k (freqs row 0 == position 4095)
    const int jf = (n & (HDIM - 1)) >> 1;
    const float cv = fcos[jf], sv = fsin[jf];
    const bool even = (n & 1) == 0;
#pragma unroll
    for (int j = 0; j < 8; ++j) {
      const float x = c[j];
      const float p = __shfl_xor(x, 1, 32);  // partner of the rotation pair
      c[j] = even ? (x * cv - p * sv) : (p * sv + x * cv);
    }
  }
  if (lane < 16) {
#pragma unroll
    for (int j = 0; j < 8; ++j) O[(size_t)j * HID + n] = c[j];
  }
}

// ----- Kernel 2: flash-decode partials ---------------------------------------
// 1024 blocks (b*128 + h*4 + part) x 256 threads (8 waves).
// Wave w handles t = part*1024 + w, step 8. Lane owns dims [4*lane, 4*lane+4).
__global__ __launch_bounds__(256) void attn_partial_kernel(
    const float* __restrict__ q_ws, const float* __restrict__ k_ws,
    const float* __restrict__ v_ws, const float* __restrict__ cache_k,
    const float* __restrict__ cache_v, const float* __restrict__ mask,
    float* __restrict__ Pm, float* __restrict__ Pl, float* __restrict__ Pacc) {
  const int blk  = blockIdx.x;
  const int part = blk & (NPART - 1);
  const int h    = (blk >> 2) & (NHEAD - 1);
  const int b    = blk >> 7;
  const int wave = threadIdx.x >> 5;
  const int lane = threadIdx.x & 31;

  const size_t bh_off = (size_t)b * HID + (size_t)h * HDIM;
  const v4f q4 = *(const v4f*)(q_ws + bh_off + lane * 4);

  float m = -INFINITY, l = 0.0f;
  v4f acc = (v4f)0.0f;

  const int tend = part * TPART + TPART;
#pragma unroll 2
  for (int t = part * TPART + wave; t < tend; t += 8) {
    const float* kp;
    const float* vp;
    if (t == NEWPOS) {  // freshly projected K/V (inputs are immutable)
      kp = k_ws + bh_off;
      vp = v_ws + bh_off;
    } else {            // cache layout (B, MAXS, H, D): streamed once -> NT
      const size_t off = (((size_t)b * SEQ + t) * NHEAD + h) * HDIM;
      kp = cache_k + off;
      vp = cache_v + off;
    }
    const v4f k4 = __builtin_nontemporal_load((const v4f*)(kp + lane * 4));
    const v4f v4 = __builtin_nontemporal_load((const v4f*)(vp + lane * 4));
    float s = q4.x * k4.x + q4.y * k4.y + q4.z * k4.z + q4.w * k4.w;
    s += __shfl_xor(s, 16, 32);
    s += __shfl_xor(s, 8, 32);
    s += __shfl_xor(s, 4, 32);
    s += __shfl_xor(s, 2, 32);
    s += __shfl_xor(s, 1, 32);
    s = s * 0.08838834764831845f + mask[t];  // 1/sqrt(128), mask (1,1,1,4096)

    const float nm = fmaxf(m, s);
    const float sc = __expf(m - nm);   // exp(-inf)=0 on first iteration
    const float pr = __expf(s - nm);
    l = l * sc + pr;
    acc = acc * sc + pr * v4;
    m = nm;
  }

  __shared__ float sm[8], sl[8];
  __shared__ float sacc[8][HDIM];
  if (lane == 0) { sm[wave] = m; sl[wave] = l; }  // m,l uniform across lanes
  *(v4f*)&sacc[wave][lane * 4] = acc;
  __syncthreads();

  if (threadIdx.x < HDIM) {
    const int d = threadIdx.x;
    float M = sm[0];
#pragma unroll
    for (int w = 1; w < 8; ++w) M = fmaxf(M, sm[w]);
    float L = 0.f, val = 0.f;
#pragma unroll
    for (int w = 0; w < 8; ++w) {
      const float e = __expf(sm[w] - M);
      L += sl[w] * e;
      val += e * sacc[w][d];
    }
    const size_t pi = ((size_t)b * NHEAD + h) * NPART + part;
    if (d == 0) { Pm[pi] = M; Pl[pi] = L; }
    Pacc[pi * HDIM + d] = val;
  }
}

// ----- Kernel 3: combine the 4 sequence partitions (log-sum-exp merge) -------
__global__ __launch_bounds__(HDIM) void attn_combine_kernel(
    const float* __restrict__ Pm, const float* __restrict__ Pl,
    const float* __restrict__ Pacc, float* __restrict__ attn_ws) {
  const int bh = blockIdx.x;     // b*32 + h
  const int d  = threadIdx.x;
  const size_t base = (size_t)bh * NPART;
  float M = Pm[base];
#pragma unroll
  for (int p = 1; p < NPART; ++p) M = fmaxf(M, Pm[base + p]);
  float L = 0.f, val = 0.f;
#pragma unroll
  for (int p = 0; p < NPART; ++p) {
    const float e = __expf(Pm[base + p] - M);
    L += Pl[base + p] * e;
    val += e * Pacc[(base + p) * HDIM + d];
  }
  attn_ws[(size_t)bh * HDIM + d] = val / L;  // == [b*4096 + h*128 + d]
}

// ----- Kernel 4: output projection out = attn @ wo.T + bo --------------------
__global__ __launch_bounds__(32) void out_proj_kernel(
    const float* __restrict__ attn_ws, const float* __restrict__ wo,
    const float* __restrict__ bo, float* __restrict__ out) {
  const int n0 = blockIdx.x * 16;
  const int lane = threadIdx.x;
  v8f c = gemm_tile16(attn_ws, wo, n0, lane);
  const int n = n0 + (lane & 15);
  const float bias = bo[n];
  if (lane < 16) {
#pragma unroll
    for (int j = 0; j < 8; ++j) out[(size_t)j * HID + n] = c[j] + bias;
  }
}

// ---------------------------------------------------------------------------
extern "C" void kernel_launch(void* const* d_in, const int* in_sizes, int n_in,
                              void* d_out, int out_size, void* d_ws,
                              size_t ws_size, hipStream_t stream) {
  const float* query    = (const float*)d_in[0];
  const float* key_in   = (const float*)d_in[1];
  const float* value_in = (const float*)d_in[2];
  const float* wq       = (const float*)d_in[3];
  const float* bq       = (const float*)d_in[4];
  const float* wk       = (const float*)d_in[5];
  const float* bk       = (const float*)d_in[6];
  const float* wv       = (const float*)d_in[7];
  const float* bv       = (const float*)d_in[8];
  const float* wo       = (const float*)d_in[9];
  const float* bo       = (const float*)d_in[10];
  const float* cache_k  = (const float*)d_in[11];
  const float* cache_v  = (const float*)d_in[12];
  const float* fcos     = (const float*)d_in[13];
  const float* fsin     = (const float*)d_in[14];
  const float* mask     = (const float*)d_in[15];
  float* out = (float*)d_out;

  // workspace carve (floats): 4x(8*4096) + 2*1024 + 1024*128  ~= 1.03 MB
  float* q_ws    = (float*)d_ws;
  float* k_ws    = q_ws + NB * HID;
  float* v_ws    = k_ws + NB * HID;
  float* attn_ws = v_ws + NB * HID;
  float* Pm      = attn_ws + NB * HID;
  float* Pl      = Pm + NB * NHEAD * NPART;
  float* Pacc    = Pl + NB * NHEAD * NPART;

  qkv_rope_kernel<<<dim3(3 * 256), dim3(32), 0, stream>>>(
      query, key_in, value_in, wq, bq, wk, bk, wv, bv, fcos, fsin,
      q_ws, k_ws, v_ws);

  attn_partial_kernel<<<dim3(NB * NHEAD * NPART), dim3(256), 0, stream>>>(
      q_ws, k_ws, v_ws, cache_k, cache_v, mask, Pm, Pl, Pacc);

  attn_combine_kernel<<<dim3(NB * NHEAD), dim3(HDIM), 0, stream>>>(
      Pm, Pl, Pacc, attn_ws);

  out_proj_kernel<<<dim3(256), dim3(32), 0, stream>>>(attn_ws, wo, bo, out);
}